// down_48309792145532
// MI455X (gfx1250) — compile-verified
//
#include <hip/hip_runtime.h>

// -------- CDNA5 WMMA types --------
typedef __attribute__((ext_vector_type(2))) float v2f;
typedef __attribute__((ext_vector_type(8))) float v8f;

#define NB     8
#define N_PTS  4096
#define NC     128
#define K_NN   16
#define NDS    2048
#define ROWS_PER_BLOCK 256
#define PSTRIDE 17   // panel stride (floats) to avoid LDS bank conflicts

// ============================================================================
// Kernel 1: per-point saliency score via WMMA-generated distance tiles.
//   d2[n][m] = sq[n] + sq[m] - 2*dot(x_n, x_m)
// One V_WMMA_F32_16X16X4_F32 per 16x16 tile:
//   A[n][k] = (-2x0, -2x1, -2x2, 1),  B[k][m] = (x0, x1, x2, sq_m),
//   C       = sq_n broadcast per row   =>   D = d2 tile directly.
// Each thread then owns one row's register-resident top-16 selection.
// ============================================================================
__global__ __launch_bounds__(256) void score_kernel(const float* __restrict__ x,
                                                    float* __restrict__ score) {
  const int b       = blockIdx.x >> 4;           // 16 row-groups per batch
  const int rowBase = (blockIdx.x & 15) * ROWS_PER_BLOCK;
  const int t       = threadIdx.x;

  __shared__ float4 col[N_PTS];                              // 64 KB: (x0,x1,x2,sq)
  __shared__ float  panel[ROWS_PER_BLOCK][PSTRIDE];          // ~17 KB d2 staging

  const float* xb = x + (size_t)b * 3 * N_PTS;
  // ---- stage all columns of this batch in LDS ----
  for (int m = t; m < N_PTS; m += 256) {
    float x0 = xb[m], x1 = xb[N_PTS + m], x2 = xb[2 * N_PTS + m];
    col[m] = make_float4(x0, x1, x2, x0 * x0 + x1 * x1 + x2 * x2);
  }
  __syncthreads();

  const int wave = t >> 5;
  const int lane = t & 31;
  const int half = lane >> 4;     // 16x4 f32 A layout: lanes 0-15 hold K=0,1; 16-31 hold K=2,3
  const int ln16 = lane & 15;

  // ---- loop-invariant A operands for this wave's two 16-row tiles ----
  const int n0 = rowBase + wave * 32 + ln16;
  const int n1 = n0 + 16;
  float4 r0 = col[n0], r1 = col[n1];
  v2f a0, a1;
  if (half == 0) {
    a0 = (v2f){-2.0f * r0.x, -2.0f * r0.y};
    a1 = (v2f){-2.0f * r1.x, -2.0f * r1.y};
  } else {
    a0 = (v2f){-2.0f * r0.z, 1.0f};
    a1 = (v2f){-2.0f * r1.z, 1.0f};
  }
  // ---- loop-invariant C operands: sq of each output row ----
  v8f c0, c1;
#pragma unroll
  for (int r = 0; r < 8; ++r) {
    c0[r] = col[rowBase + wave * 32 + r + 8 * half].w;
    c1[r] = col[rowBase + wave * 32 + 16 + r + 8 * half].w;
  }

  // ---- register-resident top-16 (sorted ascending) ----
  float nn_d[K_NN];
  int   nn_i[K_NN];
#pragma unroll
  for (int i = 0; i < K_NN; ++i) { nn_d[i] = __builtin_inff(); nn_i[i] = 0; }

  for (int mt = 0; mt < N_PTS / 16; ++mt) {
    const int colBase = mt * 16;
    float4 f = col[colBase + ln16];
    v2f bv = (half == 0) ? (v2f){f.x, f.y} : (v2f){f.z, f.w};

    // D = A*B + C  ->  16x16 tile of d2
    v8f d0 = __builtin_amdgcn_wmma_f32_16x16x4_f32(false, a0, false, bv,
                                                   (short)0, c0, false, false);
    v8f d1 = __builtin_amdgcn_wmma_f32_16x16x4_f32(false, a1, false, bv,
                                                   (short)0, c1, false, false);
#pragma unroll
    for (int r = 0; r < 8; ++r) {
      panel[wave * 32 + r + 8 * half][ln16]      = d0[r];
      panel[wave * 32 + 16 + r + 8 * half][ln16] = d1[r];
    }
    __syncthreads();

    // thread t consumes panel row t: streaming sorted-insert of 16 candidates
#pragma unroll
    for (int j = 0; j < 16; ++j) {
      float d = panel[t][j];
      int   m = colBase + j;
      if (d < nn_d[K_NN - 1]) {            // strict <  => lowest-index tie-break
#pragma unroll
        for (int i = K_NN - 1; i > 0; --i) {
          float keep = nn_d[i]; int keepi = nn_i[i];
          bool shift = d < nn_d[i - 1];
          bool here  = d < keep;
          nn_d[i] = shift ? nn_d[i - 1] : (here ? d : keep);
          nn_i[i] = shift ? nn_i[i - 1] : (here ? m : keepi);
        }
        bool h0 = d < nn_d[0];
        nn_i[0] = h0 ? m : nn_i[0];
        nn_d[0] = h0 ? d : nn_d[0];
      }
    }
    __syncthreads();
  }

  // ---- recompute score exactly like the reference: sum_k ||x_m - x_n||^2 ----
  float4 me = col[rowBase + t];
  float s = 0.0f;
#pragma unroll
  for (int j = 0; j < K_NN; ++j) {
    float4 g = col[nn_i[j]];
    float dx = g.x - me.x, dy = g.y - me.y, dz = g.z - me.z;
    s += dx * dx + dy * dy + dz * dz;
  }
  score[b * N_PTS + rowBase + t] = s;
}

// ============================================================================
// Kernel 2: per-batch bitonic sort of (score, idx), descending score with
// lax.top_k tie-break (lower index first); emit first NDS indices.
// ============================================================================
__global__ __launch_bounds__(1024) void sort_kernel(const float* __restrict__ score,
                                                    int* __restrict__ idx_out) {
  const int b = blockIdx.x;
  const int t = threadIdx.x;
  __shared__ float s[N_PTS];
  __shared__ int   id[N_PTS];
  for (int i = t; i < N_PTS; i += 1024) { s[i] = score[b * N_PTS + i]; id[i] = i; }
  __syncthreads();

  for (int k = 2; k <= N_PTS; k <<= 1) {
    for (int j = k >> 1; j > 0; j >>= 1) {
      for (int i = t; i < N_PTS; i += 1024) {
        int ixj = i ^ j;
        if (ixj > i) {
          bool up = ((i & k) == 0);
          float si = s[i], sj = s[ixj];
          int   ii = id[i], ij = id[ixj];
          bool iBefore = (si > sj) || (si == sj && ii < ij);  // rank order
          if (iBefore != up) { s[i] = sj; s[ixj] = si; id[i] = ij; id[ixj] = ii; }
        }
      }
      __syncthreads();
    }
  }
  for (int i = t; i < NDS; i += 1024) idx_out[b * NDS + i] = id[i];
}

// ============================================================================
// Kernel 3: gather outputs. d_out = [xyz (8,3,2048) | points (8,128,2048)].
// ============================================================================
__global__ void gather_kernel(const float* __restrict__ x, const float* __restrict__ y,
                              const int* __restrict__ idx, float* __restrict__ out) {
  const long xyzTotal = (long)NB * 3 * NDS;
  const long ptsTotal = (long)NB * NC * NDS;
  long gid = (long)blockIdx.x * blockDim.x + threadIdx.x;
  if (gid < xyzTotal) {
    int b = (int)(gid / (3 * NDS));
    int rem = (int)(gid % (3 * NDS));
    int d = rem / NDS, j = rem % NDS;
    out[gid] = x[((long)b * 3 + d) * N_PTS + idx[b * NDS + j]];
  } else if (gid < xyzTotal + ptsTotal) {
    long g = gid - xyzTotal;
    int b = (int)(g / (NC * NDS));
    int rem = (int)(g % (NC * NDS));
    int c = rem / NDS, j = rem % NDS;
    out[gid] = y[((long)b * NC + c) * N_PTS + idx[b * NDS + j]];
  }
}

// ============================================================================
extern "C" void kernel_launch(void* const* d_in, const int* in_sizes, int n_in,
                              void* d_out, int out_size, void* d_ws, size_t ws_size,
                              hipStream_t stream) {
  const float* x = (const float*)d_in[0];   // (8,3,4096) f32
  const float* y = (const float*)d_in[1];   // (8,128,4096) f32
  // d_in[2] = npts_ds scalar (fixed 2048, baked into NDS)
  float* score = (float*)d_ws;                                    // 8*4096 f32
  int*   idx   = (int*)((char*)d_ws + (size_t)NB * N_PTS * sizeof(float));
  float* out   = (float*)d_out;

  score_kernel<<<dim3(NB * (N_PTS / ROWS_PER_BLOCK)), dim3(256), 0, stream>>>(x, score);
  sort_kernel<<<dim3(NB), dim3(1024), 0, stream>>>(score, idx);

  long total = (long)NB * 3 * NDS + (long)NB * NC * NDS;
  gather_kernel<<<dim3((unsigned)((total + 255) / 256)), dim3(256), 0, stream>>>(x, y, idx, out);
}